// HGLSSM_45535243272449
// MI455X (gfx1250) — compile-verified
//
#include <hip/hip_runtime.h>
#include <math.h>

typedef __attribute__((ext_vector_type(2))) float v2f;
typedef __attribute__((ext_vector_type(8))) float v8f;

#define B_N   2
#define H_N   64
#define W_N   64
#define L_N   4096      // H*W
#define C_N   192
#define DI_N  384
#define DS_N  16
#define DTR_N 12
#define DC_N  4
#define HID_N 384
#define NSEQ  128       // sequences per branch (B*W or B*H)
#define T_N   64        // sequence length
#define M_TOK 8192      // NSEQ*T_N == B_N*L_N
#define XDBL_P 48       // 44 padded to 48 for 16-wide tiles

// ---------------------------------------------------------------- indexing
__device__ __forceinline__ int branch_pos(int br, int s, int t) {
    // maps (sequence s, time t) of branch br -> flat token index b*L + h*W + w
    int b, p;
    if (br == 0)      { b = s / W_N; int w = s % W_N; p = t * W_N + w; }            // scan h fwd
    else if (br == 1) { b = s / H_N; int h = s % H_N; p = h * W_N + t; }            // scan w fwd
    else if (br == 2) { b = s / W_N; int w = s % W_N; p = (H_N - 1 - t) * W_N + w; }// scan h bwd
    else              { b = s / H_N; int hf = s % H_N; p = (H_N - 1 - hf) * W_N + t; } // flipped rows, w fwd
    return b * L_N + p;
}

// ---------------------------------------------------------------- gather x -> sequence layout
__global__ void gather_seq(const float* __restrict__ x, float* __restrict__ xs, int br) {
    int idx = blockIdx.x * blockDim.x + threadIdx.x;
    if (idx >= M_TOK * C_N) return;
    int r = idx / C_N, c = idx % C_N;
    int s = r / T_N, t = r % T_N;
    xs[idx] = x[(size_t)branch_pos(br, s, t) * C_N + c];
}

// ---------------------------------------------------------------- pad x_proj weights (44 -> 48 rows, zero fill)
__global__ void pad_xprojw(const float* __restrict__ w, float* __restrict__ wp) {
    int idx = blockIdx.x * blockDim.x + threadIdx.x;     // 4*48*384
    if (idx >= 4 * XDBL_P * DI_N) return;
    int br  = idx / (XDBL_P * DI_N);
    int rem = idx % (XDBL_P * DI_N);
    int n = rem / DI_N, k = rem % DI_N;
    wp[idx] = (n < 44) ? w[((size_t)br * 44 + n) * DI_N + k] : 0.f;
}

// ---------------------------------------------------------------- fp32 WMMA GEMM: C = A(MxK) @ W(NxK)^T
// All dimensions in-bounds by construction (weights pre-padded), so the inner
// loop is branch-free: pure global_load_b64 -> v_wmma_f32_16x16x4_f32 with
// 4 independent accumulator chains (K % 16 == 0 always here).
// mode 0: C = v (+bias)        mode 1: C = gelu(v + bias)
// mode 2: C += v + bias        mode 3: C = v + bias + resid
__global__ __launch_bounds__(256)
void gemm16(const float* __restrict__ A, const float* __restrict__ Wt,
            const float* __restrict__ bias, const float* __restrict__ resid,
            float* __restrict__ C, int N, int K, int mode) {
    const int lane = threadIdx.x & 31;
    const int wave = threadIdx.x >> 5;
    const int n0   = blockIdx.x * 16;
    const int m0   = blockIdx.y * 128 + wave * 16;
    const int half = lane >> 4;      // 0: K sub 0/1, 1: K sub 2/3 (ISA A/B frag layout)
    const int lr   = lane & 15;

    const float* ap = A  + (size_t)(m0 + lr) * K + 2 * half;
    const float* bp = Wt + (size_t)(n0 + lr) * K + 2 * half;

    v8f acc0 = {}, acc1 = {}, acc2 = {}, acc3 = {};
    for (int kk = 0; kk < K; kk += 16) {
        v2f a0 = *(const v2f*)(ap + kk);
        v2f a1 = *(const v2f*)(ap + kk + 4);
        v2f a2 = *(const v2f*)(ap + kk + 8);
        v2f a3 = *(const v2f*)(ap + kk + 12);
        v2f b0 = *(const v2f*)(bp + kk);
        v2f b1 = *(const v2f*)(bp + kk + 4);
        v2f b2 = *(const v2f*)(bp + kk + 8);
        v2f b3 = *(const v2f*)(bp + kk + 12);
        acc0 = __builtin_amdgcn_wmma_f32_16x16x4_f32(false, a0, false, b0, (short)0, acc0, false, false);
        acc1 = __builtin_amdgcn_wmma_f32_16x16x4_f32(false, a1, false, b1, (short)0, acc1, false, false);
        acc2 = __builtin_amdgcn_wmma_f32_16x16x4_f32(false, a2, false, b2, (short)0, acc2, false, false);
        acc3 = __builtin_amdgcn_wmma_f32_16x16x4_f32(false, a3, false, b3, (short)0, acc3, false, false);
    }

#pragma unroll
    for (int r = 0; r < 8; ++r) {
        int row = m0 + r + 8 * half;     // ISA C/D frag layout
        int col = n0 + lr;
        float v = (acc0[r] + acc1[r]) + (acc2[r] + acc3[r]);
        if (bias) v += bias[col];
        size_t o = (size_t)row * N + col;
        if (mode == 1)      C[o] = 0.5f * v * (1.0f + erff(v * 0.70710678118f)); // exact GELU
        else if (mode == 2) C[o] += v;
        else if (mode == 3) C[o] = v + resid[o];
        else                C[o] = v;
    }
}

// ---------------------------------------------------------------- causal depthwise conv1d + SiLU
__global__ void dwconv_silu(const float* __restrict__ xz, const float* __restrict__ cw,
                            const float* __restrict__ cb, float* __restrict__ xc) {
    int idx = blockIdx.x * blockDim.x + threadIdx.x;
    if (idx >= M_TOK * DI_N) return;
    int r = idx / DI_N, d = idx % DI_N;
    int s = r / T_N, t = r % T_N;
    float acc = cb[d];
#pragma unroll
    for (int j = 0; j < DC_N; ++j) {
        int tt = t - (DC_N - 1) + j;
        if (tt >= 0)
            acc += cw[d * DC_N + j] * xz[(size_t)(s * T_N + tt) * (2 * DI_N) + d];
    }
    xc[idx] = acc / (1.0f + __expf(-acc));   // SiLU
}

// ---------------------------------------------------------------- dt projection + softplus (K=12)
__global__ void dtproj_softplus(const float* __restrict__ xdbl, const float* __restrict__ wdt,
                                const float* __restrict__ bdt, float* __restrict__ delta) {
    int idx = blockIdx.x * blockDim.x + threadIdx.x;
    if (idx >= M_TOK * DI_N) return;
    int r = idx / DI_N, d = idx % DI_N;
    float acc = bdt[d];
#pragma unroll
    for (int j = 0; j < DTR_N; ++j)
        acc += xdbl[(size_t)r * XDBL_P + j] * wdt[d * DTR_N + j];
    delta[idx] = (acc > 20.f) ? acc : log1pf(__expf(acc));
}

// ---------------------------------------------------------------- selective scan: thread per (seq, channel)
__global__ void selective_scan(const float* __restrict__ xc, const float* __restrict__ delta,
                               const float* __restrict__ xdbl, const float* __restrict__ xz,
                               const float* __restrict__ Alog, const float* __restrict__ Dp,
                               float* __restrict__ y) {
    int idx = blockIdx.x * blockDim.x + threadIdx.x;
    if (idx >= NSEQ * DI_N) return;
    int s = idx / DI_N, d = idx % DI_N;
    float Ar[DS_N], h[DS_N];
#pragma unroll
    for (int n = 0; n < DS_N; ++n) { Ar[n] = -__expf(Alog[d * DS_N + n]); h[n] = 0.f; }
    float Dd = Dp[d];
    for (int t = 0; t < T_N; ++t) {
        int r = s * T_N + t;
        float dl = delta[(size_t)r * DI_N + d];
        float u  = xc[(size_t)r * DI_N + d];
        float du = dl * u;
        const float* Bs = xdbl + (size_t)r * XDBL_P + DTR_N;
        const float* Cs = Bs + DS_N;
        float yv = 0.f;
#pragma unroll
        for (int n = 0; n < DS_N; ++n) {
            h[n] = __expf(dl * Ar[n]) * h[n] + du * Bs[n];
            yv  += h[n] * Cs[n];
        }
        float z  = xz[(size_t)r * (2 * DI_N) + DI_N + d];
        float sz = z / (1.0f + __expf(-z));
        y[(size_t)r * DI_N + d] = (yv + u * Dd) * sz;
    }
}

// ---------------------------------------------------------------- scatter branch result * gate weight
__global__ void scatter_gate(const float* __restrict__ yout, const float* __restrict__ gate,
                             float* __restrict__ grp, int br) {
    int idx = blockIdx.x * blockDim.x + threadIdx.x;
    if (idx >= M_TOK * C_N) return;
    int r = idx / C_N, c = idx % C_N;
    int s = r / T_N, t = r % T_N;
    int p = branch_pos(br, s, t);
    int b = p / L_N;
    float g = gate[(br >= 2 ? 4 : 0) + b * 2 + (br & 1)];
    grp[(size_t)p * C_N + c] += g * yout[idx];
}

// ---------------------------------------------------------------- pooled mean + gate MLPs
__global__ void pool_mean(const float* __restrict__ x, float* __restrict__ pooled) {
    int b = blockIdx.x, c = threadIdx.x;
    float acc = 0.f;
    for (int p = 0; p < L_N; ++p) acc += x[((size_t)b * L_N + p) * C_N + c];
    pooled[b * C_N + c] = acc * (1.0f / L_N);
}

__global__ void gate_mlp(const float* __restrict__ pooled,
                         const float* __restrict__ w1, const float* __restrict__ b1,
                         const float* __restrict__ w2, const float* __restrict__ b2,
                         float* __restrict__ gate) {
    int g = blockIdx.x >> 1, b = blockIdx.x & 1;
    int j = threadIdx.x;                       // 48 threads
    __shared__ float t1[48];
    float acc = b1[g * 48 + j];
    for (int c = 0; c < C_N; ++c)
        acc += pooled[b * C_N + c] * w1[((size_t)g * 48 + j) * C_N + c];
    t1[j] = fmaxf(acc, 0.f);
    __syncthreads();
    if (j == 0) {
        float l0 = b2[g * 2 + 0], l1 = b2[g * 2 + 1];
        for (int k = 0; k < 48; ++k) {
            l0 += t1[k] * w2[(g * 2 + 0) * 48 + k];
            l1 += t1[k] * w2[(g * 2 + 1) * 48 + k];
        }
        float m = fmaxf(l0, l1);
        float e0 = __expf(l0 - m), e1 = __expf(l1 - m);
        float inv = 1.f / (e0 + e1);
        gate[g * 4 + b * 2 + 0] = e0 * inv;
        gate[g * 4 + b * 2 + 1] = e1 * inv;
    }
}

// ---------------------------------------------------------------- depthwise 3x3 SAME conv (NHWC)
__global__ void dwconv3x3(const float* __restrict__ hin, const float* __restrict__ cw,
                          const float* __restrict__ cb, float* __restrict__ hout) {
    int idx = blockIdx.x * blockDim.x + threadIdx.x;
    if (idx >= B_N * L_N * HID_N) return;
    int c = idx % HID_N;
    int r = idx / HID_N;
    int b = r / L_N; int pp = r % L_N; int h = pp / W_N; int w = pp % W_N;
    float acc = cb[c];
#pragma unroll
    for (int ky = 0; ky < 3; ++ky) {
        int hh = h + ky - 1; if (hh < 0 || hh >= H_N) continue;
#pragma unroll
        for (int kx = 0; kx < 3; ++kx) {
            int ww = w + kx - 1; if (ww < 0 || ww >= W_N) continue;
            acc += cw[(c * 3 + ky) * 3 + kx] * hin[((size_t)b * L_N + hh * W_N + ww) * HID_N + c];
        }
    }
    hout[idx] = acc;
}

__global__ void zero_buf(float* __restrict__ p, int n) {
    int i = blockIdx.x * blockDim.x + threadIdx.x;
    if (i < n) p[i] = 0.f;
}

// ================================================================ host
extern "C" void kernel_launch(void* const* d_in, const int* in_sizes, int n_in,
                              void* d_out, int out_size, void* d_ws, size_t ws_size,
                              hipStream_t stream) {
    const float* x        = (const float*)d_in[0];
    const float* m_inproj = (const float*)d_in[3];   // (4, 768, 192)
    const float* m_convw  = (const float*)d_in[4];   // (4, 384, 4)
    const float* m_convb  = (const float*)d_in[5];
    const float* m_xprojw = (const float*)d_in[6];   // (4, 44, 384)
    const float* m_dtw    = (const float*)d_in[7];   // (4, 384, 12)
    const float* m_dtb    = (const float*)d_in[8];
    const float* m_Alog   = (const float*)d_in[9];   // (4, 384, 16)
    const float* m_D      = (const float*)d_in[10];
    const float* m_outw   = (const float*)d_in[11];  // (4, 192, 384)
    const float* g_w1     = (const float*)d_in[12];
    const float* g_b1     = (const float*)d_in[13];
    const float* g_w2     = (const float*)d_in[14];
    const float* g_b2     = (const float*)d_in[15];
    const float* l_fc1w   = (const float*)d_in[16];  // (2, 384, 192)
    const float* l_fc1b   = (const float*)d_in[17];
    const float* l_convw  = (const float*)d_in[18];  // (2, 384, 3, 3)
    const float* l_convb  = (const float*)d_in[19];
    const float* l_fc2w   = (const float*)d_in[20];  // (2, 192, 384)
    const float* l_fc2b   = (const float*)d_in[21];
    const float* out_w    = (const float*)d_in[22];
    const float* out_b    = (const float*)d_in[23];
    float* out = (float*)d_out;

    float* ws = (float*)d_ws;
    size_t oXS    = 0;                               // 8192*192
    size_t oXZ    = oXS    + (size_t)M_TOK * C_N;    // 8192*768
    size_t oXC    = oXZ    + (size_t)M_TOK * 2 * DI_N;
    size_t oXDBL  = oXC    + (size_t)M_TOK * DI_N;   // 8192*48
    size_t oDELTA = oXDBL  + (size_t)M_TOK * XDBL_P;
    size_t oY     = oDELTA + (size_t)M_TOK * DI_N;
    size_t oYOUT  = oY     + (size_t)M_TOK * DI_N;
    size_t oGRPF  = oYOUT  + (size_t)M_TOK * C_N;
    size_t oGRPB  = oGRPF  + (size_t)M_TOK * C_N;
    size_t oGATE  = oGRPB  + (size_t)M_TOK * C_N;    // pooled(384) + gate(8)
    size_t oXPW   = oGATE  + 512;                    // padded x_proj weights 4*48*384
    float* XS = ws + oXS;   float* XZ = ws + oXZ;    float* XC = ws + oXC;
    float* XDBL = ws + oXDBL; float* DELTA = ws + oDELTA; float* Y = ws + oY;
    float* YOUT = ws + oYOUT; float* GRPF = ws + oGRPF; float* GRPB = ws + oGRPB;
    float* POOLED = ws + oGATE; float* GATE = POOLED + B_N * C_N;
    float* XPW = ws + oXPW;
    // LeFF aliases over freed mamba buffers
    float* HB   = XC;      // 8192*384
    float* CONV = DELTA;   // 8192*384
    float* ZSUM = Y;       // 8192*192 (fits in 8192*384)

    const int TPB = 256;
    const int nTokC  = M_TOK * C_N;
    const int nTokDI = M_TOK * DI_N;

    // pre-pad x_proj weights once (44 -> 48 rows, zero fill)
    pad_xprojw<<<(4 * XDBL_P * DI_N + TPB - 1) / TPB, TPB, 0, stream>>>(m_xprojw, XPW);

    // gate weights
    pool_mean<<<B_N, C_N, 0, stream>>>(x, POOLED);
    gate_mlp<<<4, 48, 0, stream>>>(POOLED, g_w1, g_b1, g_w2, g_b2, GATE);

    // zero accumulators (GRPF/GRPB contiguous)
    zero_buf<<<(2 * nTokC + TPB - 1) / TPB, TPB, 0, stream>>>(GRPF, 2 * nTokC);

    for (int br = 0; br < 4; ++br) {
        gather_seq<<<(nTokC + TPB - 1) / TPB, TPB, 0, stream>>>(x, XS, br);
        gemm16<<<dim3(768 / 16, M_TOK / 128), TPB, 0, stream>>>(
            XS, m_inproj + (size_t)br * 768 * C_N, nullptr, nullptr, XZ, 768, C_N, 0);
        dwconv_silu<<<(nTokDI + TPB - 1) / TPB, TPB, 0, stream>>>(
            XZ, m_convw + (size_t)br * DI_N * DC_N, m_convb + br * DI_N, XC);
        gemm16<<<dim3(XDBL_P / 16, M_TOK / 128), TPB, 0, stream>>>(
            XC, XPW + (size_t)br * XDBL_P * DI_N, nullptr, nullptr, XDBL, XDBL_P, DI_N, 0);
        dtproj_softplus<<<(nTokDI + TPB - 1) / TPB, TPB, 0, stream>>>(
            XDBL, m_dtw + (size_t)br * DI_N * DTR_N, m_dtb + br * DI_N, DELTA);
        selective_scan<<<(NSEQ * DI_N + TPB - 1) / TPB, TPB, 0, stream>>>(
            XC, DELTA, XDBL, XZ, m_Alog + (size_t)br * DI_N * DS_N, m_D + br * DI_N, Y);
        gemm16<<<dim3(C_N / 16, M_TOK / 128), TPB, 0, stream>>>(
            Y, m_outw + (size_t)br * C_N * DI_N, nullptr, nullptr, YOUT, C_N, DI_N, 0);
        scatter_gate<<<(nTokC + TPB - 1) / TPB, TPB, 0, stream>>>(
            YOUT, GATE, (br < 2) ? GRPF : GRPB, br);
    }

    // LeFF blocks
    zero_buf<<<(nTokC + TPB - 1) / TPB, TPB, 0, stream>>>(ZSUM, nTokC);
    for (int g = 0; g < 2; ++g) {
        const float* grp = (g == 0) ? GRPF : GRPB;
        gemm16<<<dim3(HID_N / 16, M_TOK / 128), TPB, 0, stream>>>(
            grp, l_fc1w + (size_t)g * HID_N * C_N, l_fc1b + g * HID_N, nullptr,
            HB, HID_N, C_N, 1 /*gelu*/);
        dwconv3x3<<<(B_N * L_N * HID_N + TPB - 1) / TPB, TPB, 0, stream>>>(
            HB, l_convw + (size_t)g * HID_N * 9, l_convb + g * HID_N, CONV);
        gemm16<<<dim3(C_N / 16, M_TOK / 128), TPB, 0, stream>>>(
            CONV, l_fc2w + (size_t)g * C_N * HID_N, l_fc2b + g * C_N, nullptr,
            ZSUM, C_N, HID_N, 2 /*accumulate*/);
    }

    // final projection + residual
    gemm16<<<dim3(C_N / 16, M_TOK / 128), TPB, 0, stream>>>(
        ZSUM, out_w, out_b, x, out, C_N, C_N, 3 /*+bias+resid*/);
}